// NeuralNDCGLoss_89352499626259
// MI455X (gfx1250) — compile-verified
//
#include <hip/hip_runtime.h>
#include <hip/hip_bf16.h>

// NeuralNDCG, N=4096, no padding (target in [0,1) never hits -1e32).
// Strategy: factor Sinkhorn as diagonal scalings around a FIXED matrix
// E[r][i] = exp(pred[i]*(4095-2r) - B[i] - rowmax_r)  (64MB, lives in L2).
// Each of the 50 iterations = two matvecs vs E. Column-direction matvec uses
// V_WMMA_F32_16X16X4_F32 (A = rowscale replicated over M, B = 4x16 E tile,
// coalesced row-major loads); row-direction matvec uses coalesced VALU + shfl.

constexpr int LN = 4096;
constexpr float SEPS = 1e-10f;
constexpr int SINK_ITERS = 50;

typedef float v2f __attribute__((ext_vector_type(2)));
typedef float v8f __attribute__((ext_vector_type(8)));

// ---- B[i] = sum_j |pred[i] - pred[j]| -------------------------------------
__global__ void kB(const float* __restrict__ pred, float* __restrict__ Bsum) {
  __shared__ float pj[256];
  int i = blockIdx.x * 256 + threadIdx.x;
  float pi = pred[i];
  float s = 0.f;
  for (int j0 = 0; j0 < LN; j0 += 256) {
    __syncthreads();
    pj[threadIdx.x] = pred[j0 + threadIdx.x];
    __syncthreads();
#pragma unroll 8
    for (int j = 0; j < 256; ++j) s += fabsf(pi - pj[j]);
  }
  Bsum[i] = s;
}

// ---- E row build: E[r][i] = exp(val - rowmax), rowscale[r] = 1/Z ----------
__global__ void kSoftmax(const float* __restrict__ pred, const float* __restrict__ Bsum,
                         float* __restrict__ E, float* __restrict__ rowscale,
                         float* __restrict__ colscale) {
  __shared__ float red[256];
  int r = blockIdx.x;
  int tid = threadIdx.x;
  float srow = (float)(LN - 1 - 2 * r);  // scaling[r] = n+1-2(r+1)
  float vals[16];
  float mx = -3.4e38f;
#pragma unroll
  for (int t = 0; t < 16; ++t) {
    int i = tid + 256 * t;
    float v = pred[i] * srow - Bsum[i];
    vals[t] = v;
    mx = fmaxf(mx, v);
  }
  red[tid] = mx; __syncthreads();
  for (int off = 128; off > 0; off >>= 1) {
    if (tid < off) red[tid] = fmaxf(red[tid], red[tid + off]);
    __syncthreads();
  }
  float rmax = red[0]; __syncthreads();
  float sum = 0.f;
#pragma unroll
  for (int t = 0; t < 16; ++t) {
    int i = tid + 256 * t;
    float e = expf(vals[t] - rmax);
    E[(size_t)r * LN + i] = e;
    sum += e;
  }
  red[tid] = sum; __syncthreads();
  for (int off = 128; off > 0; off >>= 1) {
    if (tid < off) red[tid] += red[tid + off];
    __syncthreads();
  }
  if (tid == 0) {
    rowscale[r] = 1.0f / red[0];  // softmax normalization folded into R
    colscale[r] = 1.0f;
  }
}

// ---- WMMA column pass: wpart[slab][i] = sum_{r in slab} R[r]*E[r][i] ------
// grid (16 col-tiles, 16 row-slabs), 256 thr = 8 waves, 32 cols/wave.
__global__ void kColPass(const float* __restrict__ E, const float* __restrict__ rowscale,
                         float* __restrict__ wpart) {
  __shared__ float rs[256];
  int tid = threadIdx.x;
  int slab = blockIdx.y;
  int rbase = slab * 256;
  rs[tid] = rowscale[rbase + tid];
  __syncthreads();
  int lane = tid & 31;
  int wave = tid >> 5;
  int colBase = blockIdx.x * 256 + wave * 32;
  int c0 = colBase + (lane & 15);
  int c1 = c0 + 16;
  int ksel = (lane & 16) ? 2 : 0;  // lanes 0-15 hold K={0,1}, 16-31 hold K={2,3}
  v8f acc0 = {};
  v8f acc1 = {};
  for (int k = 0; k < 64; ++k) {
    int lr = 4 * k + ksel;
    v2f a; a[0] = rs[lr]; a[1] = rs[lr + 1];          // A[m][k]=R[k] for all m
    const float* row0 = E + (size_t)(rbase + lr) * LN;
    const float* row1 = row0 + LN;
    v2f b0; b0[0] = row0[c0]; b0[1] = row1[c0];        // B[k][n]=E[k][n]
    v2f b1; b1[0] = row0[c1]; b1[1] = row1[c1];
    acc0 = __builtin_amdgcn_wmma_f32_16x16x4_f32(false, a, false, b0,
                                                 (short)0, acc0, false, false);
    acc1 = __builtin_amdgcn_wmma_f32_16x16x4_f32(false, a, false, b1,
                                                 (short)0, acc1, false, false);
  }
  // All D rows identical (A constant over M). VGPR0: lanes0-15 -> M=0,N=lane;
  // lanes16-31 -> M=8,N=lane-16. Write 32 cols per wave.
  float outv = (lane < 16) ? acc0[0] : acc1[0];
  wpart[slab * LN + colBase + lane] = outv;
}

// ---- colscale update: C[i] = C[i] / max(C[i]*w[i], eps) -------------------
__global__ void kColUpdate(const float* __restrict__ wpart, float* __restrict__ colscale) {
  int i = blockIdx.x * 256 + threadIdx.x;
  float w = 0.f;
#pragma unroll
  for (int s = 0; s < 16; ++s) w += wpart[s * LN + i];
  float c = colscale[i];
  colscale[i] = c / fmaxf(c * w, SEPS);
}

// ---- row pass + fused rowscale update: R[r] = R[r]/max(R[r]*(E·C)[r],eps) -
__global__ void kRowPass(const float* __restrict__ E, const float* __restrict__ colscale,
                         float* __restrict__ rowscale) {
  __shared__ float cs[LN];
  int tid = threadIdx.x;
#pragma unroll
  for (int t = 0; t < 16; ++t) cs[tid + 256 * t] = colscale[tid + 256 * t];
  __syncthreads();
  int lane = tid & 31, wave = tid >> 5;
  int r = blockIdx.x * 8 + wave;
  const float* row = E + (size_t)r * LN;
  float sum = 0.f;
  for (int t = 0; t < 128; ++t) {
    int i = lane + 32 * t;
    sum += row[i] * cs[i];
  }
  for (int off = 16; off > 0; off >>= 1) sum += __shfl_xor(sum, off, 32);
  if (lane == 0) {
    float Rr = rowscale[r];
    rowscale[r] = Rr / fmaxf(Rr * sum, SEPS);
  }
}

// ---- numerator: sum_r disc[r]*R[r]*(E·(C⊙gains))[r] -> npart[block] -------
__global__ void kGroundTruth(const float* __restrict__ E, const float* __restrict__ colscale,
                             const float* __restrict__ rowscale, const float* __restrict__ target,
                             float* __restrict__ npart) {
  __shared__ float q[LN];
  __shared__ float wv[8];
  int tid = threadIdx.x;
#pragma unroll
  for (int t = 0; t < 16; ++t) {
    int i = tid + 256 * t;
    q[i] = colscale[i] * (exp2f(target[i]) - 1.0f);
  }
  __syncthreads();
  int lane = tid & 31, wave = tid >> 5;
  int r = blockIdx.x * 8 + wave;
  const float* row = E + (size_t)r * LN;
  float sum = 0.f;
  for (int t = 0; t < 128; ++t) {
    int i = lane + 32 * t;
    sum += row[i] * q[i];
  }
  for (int off = 16; off > 0; off >>= 1) sum += __shfl_xor(sum, off, 32);
  if (lane == 0) {
    float disc = 1.0f / log2f((float)r + 2.0f);
    wv[wave] = disc * rowscale[r] * sum;
  }
  __syncthreads();
  if (tid == 0) {
    float s = 0.f;
    for (int w = 0; w < 8; ++w) s += wv[w];
    npart[blockIdx.x] = s;
  }
}

// ---- iDCG via stable-descending rank-by-counting --------------------------
__global__ void kIdcg(const float* __restrict__ target, float* __restrict__ ipart) {
  __shared__ float tg[LN];
  __shared__ float red[256];
  int tid = threadIdx.x;
#pragma unroll
  for (int t = 0; t < 16; ++t) tg[tid + 256 * t] = target[tid + 256 * t];
  __syncthreads();
  int i = blockIdx.x * 256 + tid;
  float ti = tg[i];
  int rank = 0;
  for (int j = 0; j < LN; ++j) {
    float tj = tg[j];
    rank += (tj > ti) ? 1 : 0;
    rank += (tj == ti && j < i) ? 1 : 0;
  }
  red[tid] = (exp2f(ti) - 1.0f) / log2f((float)rank + 2.0f);
  __syncthreads();
  for (int off = 128; off > 0; off >>= 1) {
    if (tid < off) red[tid] += red[tid + off];
    __syncthreads();
  }
  if (tid == 0) ipart[blockIdx.x] = red[0];
}

// ---- finalize: loss = -(num / (idcg + 1e-8)) ------------------------------
__global__ void kFinal(const float* __restrict__ npart, const float* __restrict__ ipart,
                       float* __restrict__ out) {
  float num = 0.f;
  for (int i = 0; i < 512; ++i) num += npart[i];
  float idcg = 0.f;
  for (int i = 0; i < 16; ++i) idcg += ipart[i];
  out[0] = (idcg == 0.0f) ? 0.0f : -(num / (idcg + 1e-8f));
}

extern "C" void kernel_launch(void* const* d_in, const int* in_sizes, int n_in,
                              void* d_out, int out_size, void* d_ws, size_t ws_size,
                              hipStream_t stream) {
  (void)in_sizes; (void)n_in; (void)out_size; (void)ws_size;
  const float* pred   = (const float*)d_in[0];
  const float* target = (const float*)d_in[1];
  float* out = (float*)d_out;

  // Workspace layout (floats): needs ~64.3 MiB
  float* ws       = (float*)d_ws;
  float* E        = ws;                                // 4096*4096
  float* Bsum     = E + (size_t)LN * LN;               // 4096
  float* rowscale = Bsum + LN;                         // 4096
  float* colscale = rowscale + LN;                     // 4096
  float* wpart    = colscale + LN;                     // 16*4096
  float* npart    = wpart + 16 * LN;                   // 512
  float* ipart    = npart + 512;                       // 16

  kB<<<LN / 256, 256, 0, stream>>>(pred, Bsum);
  kSoftmax<<<LN, 256, 0, stream>>>(pred, Bsum, E, rowscale, colscale);

  for (int it = 0; it < SINK_ITERS; ++it) {
    kColPass<<<dim3(16, 16), 256, 0, stream>>>(E, rowscale, wpart);
    kColUpdate<<<LN / 256, 256, 0, stream>>>(wpart, colscale);
    kRowPass<<<LN / 8, 256, 0, stream>>>(E, colscale, rowscale);
  }

  kGroundTruth<<<LN / 8, 256, 0, stream>>>(E, colscale, rowscale, target, npart);
  kIdcg<<<LN / 256, 256, 0, stream>>>(target, ipart);
  kFinal<<<1, 1, 0, stream>>>(npart, ipart, out);
}